// LightGCN_49675591745621
// MI455X (gfx1250) — compile-verified
//
#include <hip/hip_runtime.h>
#include <cstdint>
#include <cstddef>

#ifndef __has_builtin
#define __has_builtin(x) 0
#endif

// ---------------------------------------------------------------------------
// CDNA5 async global->LDS helpers (gfx1250). Builtin if present, asm fallback.
// ---------------------------------------------------------------------------
typedef __attribute__((address_space(3))) int  lgcn_as3_i32;
typedef __attribute__((address_space(1))) int  lgcn_as1_i32;

__device__ __forceinline__ void lgcn_async_b32(const int* gp, int* lp) {
#if __has_builtin(__builtin_amdgcn_global_load_async_to_lds_b32)
  __builtin_amdgcn_global_load_async_to_lds_b32((lgcn_as1_i32*)const_cast<int*>(gp),
                                                (lgcn_as3_i32*)lp, 0, 0);
#else
  asm volatile("global_load_async_to_lds_b32 %0, %1, off"
               :
               : "v"((unsigned)(uintptr_t)lp), "v"(gp)
               : "memory");
#endif
}

__device__ __forceinline__ void lgcn_wait_async0() {
#if __has_builtin(__builtin_amdgcn_s_wait_asynccnt)
  __builtin_amdgcn_s_wait_asynccnt(0);
#else
  asm volatile("s_wait_asynccnt 0" ::: "memory");
#endif
  asm volatile("" ::: "memory");
}

__device__ __forceinline__ void lgcn_wait_async1() {
#if __has_builtin(__builtin_amdgcn_s_wait_asynccnt)
  __builtin_amdgcn_s_wait_asynccnt(1);
#else
  asm volatile("s_wait_asynccnt 1" ::: "memory");
#endif
  asm volatile("" ::: "memory");
}

// ---------------------------------------------------------------------------
// Setup kernels
// ---------------------------------------------------------------------------
__global__ __launch_bounds__(256) void k_zero(int* __restrict__ p, int nwords) {
  int i = blockIdx.x * 256 + threadIdx.x;
  if (i < nwords) p[i] = 0;
}

__global__ __launch_bounds__(256) void k_degree(const int* __restrict__ src,
                                                const int* __restrict__ dst,
                                                int* __restrict__ deg_out,
                                                int* __restrict__ deg_in,
                                                int e) {
  int i = blockIdx.x * 256 + threadIdx.x;
  if (i < e) {
    atomicAdd(&deg_out[src[i]], 1);
    atomicAdd(&deg_in[dst[i]], 1);
  }
}

__global__ __launch_bounds__(256) void k_scale(const int* __restrict__ deg_out,
                                               const int* __restrict__ deg_in,
                                               float* __restrict__ scale_out,
                                               float* __restrict__ scale_in,
                                               const int* __restrict__ num_layers,
                                               int n) {
  int i = blockIdx.x * 256 + threadIdx.x;
  if (i < n) {
    int L = num_layers[0];
    float alpha = 0.0f;
    for (int k = 0; k < L; ++k) alpha += 1.0f / (1.0f + (float)k);
    int d_o = deg_out[i]; if (d_o < 1) d_o = 1;
    int d_i = deg_in[i];  if (d_i < 1) d_i = 1;
    scale_out[i] = 1.0f / sqrtf((float)d_o);
    scale_in[i]  = alpha / sqrtf((float)d_i);
  }
}

// h[i,:] = emb[i,:] * scale_out[i]   (one thread per float4, coalesced)
__global__ __launch_bounds__(256) void k_prescale(const float* __restrict__ emb,
                                                  const float* __restrict__ scale_out,
                                                  float* __restrict__ h,
                                                  int n) {
  int t = blockIdx.x * 256 + threadIdx.x;   // n*16 threads, 4 floats each
  if (t < n * 16) {
    int node = t >> 4;
    float sc = scale_out[node];
    float4 v = *(const float4*)(emb + (size_t)t * 4);
    v.x *= sc; v.y *= sc; v.z *= sc; v.w *= sc;
    *(float4*)(h + (size_t)t * 4) = v;
  }
}

// ---------------------------------------------------------------------------
// Scan (exclusive) of deg_in -> row_ptr
// ---------------------------------------------------------------------------
__global__ __launch_bounds__(256) void k_scan_block(const int* __restrict__ deg,
                                                    int* __restrict__ row_ptr,
                                                    int* __restrict__ bsums,
                                                    int n) {
  __shared__ int t[256];
  const int tid = threadIdx.x;
  const int i = blockIdx.x * 256 + tid;
  const int v = (i < n) ? deg[i] : 0;
  int sum = v;
  t[tid] = sum;
  __syncthreads();
  for (int off = 1; off < 256; off <<= 1) {
    int a = (tid >= off) ? t[tid - off] : 0;
    __syncthreads();
    sum += a;
    t[tid] = sum;
    __syncthreads();
  }
  if (i < n) row_ptr[i] = sum - v;
  if (tid == 255) bsums[blockIdx.x] = sum;
}

__global__ __launch_bounds__(1024) void k_scan_top(int* __restrict__ bsums, int b) {
  __shared__ int t[1024];
  const int tid = threadIdx.x;
  const int v = (tid < b) ? bsums[tid] : 0;
  int sum = v;
  t[tid] = sum;
  __syncthreads();
  for (int off = 1; off < 1024; off <<= 1) {
    int a = (tid >= off) ? t[tid - off] : 0;
    __syncthreads();
    sum += a;
    t[tid] = sum;
    __syncthreads();
  }
  if (tid < b) bsums[tid] = sum - v;
}

__global__ __launch_bounds__(256) void k_scan_add(int* __restrict__ row_ptr,
                                                  const int* __restrict__ bsums,
                                                  int n) {
  int i = blockIdx.x * 256 + threadIdx.x;
  if (i < n) row_ptr[i] += bsums[blockIdx.x];
}

// Build CSR over dst: csr[row_ptr[dst] + slot] = src
__global__ __launch_bounds__(256) void k_scatter(const int* __restrict__ src,
                                                 const int* __restrict__ dst,
                                                 const int* __restrict__ row_ptr,
                                                 int* __restrict__ cursor,
                                                 int* __restrict__ csr,
                                                 int e) {
  int i = blockIdx.x * 256 + threadIdx.x;
  if (i < e) {
    int d = dst[i];
    int p = atomicAdd(&cursor[d], 1);
    csr[row_ptr[d] + p] = src[i];
  }
}

// ---------------------------------------------------------------------------
// Gather: one wave32 per node, lane owns dims {2*lane, 2*lane+1}. Neighbor
// indices double-buffered in LDS via async global->LDS (s_wait_asynccnt 1
// overlaps the next chunk's fetch with the current chunk's compute).
// ---------------------------------------------------------------------------
__global__ __launch_bounds__(256) void k_gather_pre(const float* __restrict__ h,
                                                    const int* __restrict__ csr,
                                                    const int* __restrict__ row_ptr,
                                                    const int* __restrict__ deg_in,
                                                    const float* __restrict__ scale_in,
                                                    float* __restrict__ out,
                                                    int n) {
  __shared__ int s_idx[8 * 64];  // 8 waves x 2 buffers x 32
  const int lane = threadIdx.x & 31;
  const int wv = threadIdx.x >> 5;
  const int node = __builtin_amdgcn_readfirstlane((int)(blockIdx.x * 8 + wv));
  if (node >= n) return;

  const int row = __builtin_amdgcn_readfirstlane(row_ptr[node]);
  const int cnt = __builtin_amdgcn_readfirstlane(deg_in[node]);
  int* slot0 = &s_idx[wv * 64];
  int* slot1 = slot0 + 32;

  float acc0 = 0.0f, acc1 = 0.0f;
  const float* hl = h + (size_t)lane * 2;
  const int nchunks = (cnt + 31) >> 5;

  if (nchunks > 0) {
    int m0 = cnt < 32 ? cnt : 32;
    if (lane < m0) lgcn_async_b32(csr + row + lane, slot0 + lane);
    for (int c = 0; c < nchunks; ++c) {
      int* cur = (c & 1) ? slot1 : slot0;
      const int base = c << 5;
      int m = cnt - base; if (m > 32) m = 32;
      if (c + 1 < nchunks) {
        const int base2 = base + 32;
        int m2 = cnt - base2; if (m2 > 32) m2 = 32;
        int* nxt = ((c + 1) & 1) ? slot1 : slot0;
        if (lane < m2) lgcn_async_b32(csr + row + base2 + lane, nxt + lane);
        lgcn_wait_async1();  // previous chunk landed (in-order completion)
      } else {
        lgcn_wait_async0();
      }
      const int v_idx = cur[lane];  // ds_load; lanes >= m hold junk, never read
#pragma unroll 4
      for (int j = 0; j < m; ++j) {
        const int s = __builtin_amdgcn_readlane(v_idx, j);  // uniform
        const float2 ev = *(const float2*)(hl + (size_t)s * 64);
        acc0 += ev.x;
        acc1 += ev.y;
      }
    }
  }

  const float si = scale_in[node];
  float2 r;
  r.x = acc0 * si;
  r.y = acc1 * si;
  *(float2*)(out + (size_t)node * 64 + (size_t)lane * 2) = r;
}

// Fallback when ws can't hold h: gather per-edge scales lane-parallel (one
// scattered load per chunk) and broadcast both index and scale via readlane.
__global__ __launch_bounds__(256) void k_gather_scale(const float* __restrict__ emb,
                                                      const int* __restrict__ csr,
                                                      const int* __restrict__ row_ptr,
                                                      const int* __restrict__ deg_in,
                                                      const float* __restrict__ scale_out,
                                                      const float* __restrict__ scale_in,
                                                      float* __restrict__ out,
                                                      int n) {
  __shared__ int s_idx[8 * 64];
  const int lane = threadIdx.x & 31;
  const int wv = threadIdx.x >> 5;
  const int node = __builtin_amdgcn_readfirstlane((int)(blockIdx.x * 8 + wv));
  if (node >= n) return;

  const int row = __builtin_amdgcn_readfirstlane(row_ptr[node]);
  const int cnt = __builtin_amdgcn_readfirstlane(deg_in[node]);
  int* slot0 = &s_idx[wv * 64];
  int* slot1 = slot0 + 32;

  float acc0 = 0.0f, acc1 = 0.0f;
  const float* embl = emb + (size_t)lane * 2;
  const int nchunks = (cnt + 31) >> 5;

  if (nchunks > 0) {
    int m0 = cnt < 32 ? cnt : 32;
    if (lane < m0) lgcn_async_b32(csr + row + lane, slot0 + lane);
    for (int c = 0; c < nchunks; ++c) {
      int* cur = (c & 1) ? slot1 : slot0;
      const int base = c << 5;
      int m = cnt - base; if (m > 32) m = 32;
      if (c + 1 < nchunks) {
        const int base2 = base + 32;
        int m2 = cnt - base2; if (m2 > 32) m2 = 32;
        int* nxt = ((c + 1) & 1) ? slot1 : slot0;
        if (lane < m2) lgcn_async_b32(csr + row + base2 + lane, nxt + lane);
        lgcn_wait_async1();
      } else {
        lgcn_wait_async0();
      }
      const int v_idx = cur[lane];
      float sc_v = 0.0f;
      if (lane < m) sc_v = scale_out[v_idx];  // lane-parallel scattered gather
#pragma unroll 4
      for (int j = 0; j < m; ++j) {
        const int s = __builtin_amdgcn_readlane(v_idx, j);
        const float sc = __int_as_float(
            __builtin_amdgcn_readlane(__float_as_int(sc_v), j));
        const float2 ev = *(const float2*)(embl + (size_t)s * 64);
        acc0 = fmaf(sc, ev.x, acc0);
        acc1 = fmaf(sc, ev.y, acc1);
      }
    }
  }

  const float si = scale_in[node];
  float2 r;
  r.x = acc0 * si;
  r.y = acc1 * si;
  *(float2*)(out + (size_t)node * 64 + (size_t)lane * 2) = r;
}

// ---------------------------------------------------------------------------
// Host launcher
// ---------------------------------------------------------------------------
extern "C" void kernel_launch(void* const* d_in, const int* in_sizes, int n_in,
                              void* d_out, int out_size, void* d_ws, size_t ws_size,
                              hipStream_t stream) {
  const float* emb = (const float*)d_in[0];
  const int* src = (const int*)d_in[1];
  const int* dst = (const int*)d_in[2];
  const int* num_layers = (const int*)d_in[3];
  float* out = (float*)d_out;

  const int n = in_sizes[0] / 64;  // EMB_DIM = 64
  const int e = in_sizes[1];
  if (n <= 0 || e <= 0) return;

  char* w = (char*)d_ws;
  auto take = [&](size_t bytes) -> char* {
    char* p = w;
    w += (bytes + 255) & ~(size_t)255;
    return p;
  };
  int* deg_out = (int*)take((size_t)n * 4);
  int* deg_in = (int*)take((size_t)n * 4);
  int* cursor = (int*)take((size_t)n * 4);
  int* row_ptr = (int*)take((size_t)(n + 1) * 4);
  int* bsums = (int*)take((size_t)1024 * 4);
  float* scale_out = (float*)take((size_t)n * 4);
  float* scale_in = (float*)take((size_t)n * 4);
  int* csr = (int*)take((size_t)e * 4);
  float* h = (float*)take((size_t)n * 64 * 4);
  const bool use_pre = ((size_t)(w - (char*)d_ws) <= ws_size);

  const int B = (n + 255) / 256;  // scan blocks (391 for N=100K, <=1024 required)
  const int zero_words = (int)(((char*)cursor + (size_t)n * 4 - (char*)deg_out) / 4);

  k_zero<<<(zero_words + 255) / 256, 256, 0, stream>>>(deg_out, zero_words);
  k_degree<<<(e + 255) / 256, 256, 0, stream>>>(src, dst, deg_out, deg_in, e);
  k_scale<<<B, 256, 0, stream>>>(deg_out, deg_in, scale_out, scale_in, num_layers, n);
  k_scan_block<<<B, 256, 0, stream>>>(deg_in, row_ptr, bsums, n);
  k_scan_top<<<1, 1024, 0, stream>>>(bsums, B);
  k_scan_add<<<B, 256, 0, stream>>>(row_ptr, bsums, n);
  k_scatter<<<(e + 255) / 256, 256, 0, stream>>>(src, dst, row_ptr, cursor, csr, e);

  if (use_pre) {
    k_prescale<<<(n * 16 + 255) / 256, 256, 0, stream>>>(emb, scale_out, h, n);
    k_gather_pre<<<(n + 7) / 8, 256, 0, stream>>>(h, csr, row_ptr, deg_in,
                                                  scale_in, out, n);
  } else {
    k_gather_scale<<<(n + 7) / 8, 256, 0, stream>>>(emb, csr, row_ptr, deg_in,
                                                    scale_out, scale_in, out, n);
  }
}